// GATClassifier_68616397521376
// MI455X (gfx1250) — compile-verified
//
#include <hip/hip_runtime.h>
#include <hip/hip_bf16.h>

// ---------------------------------------------------------------------------
// GATv2 (2 layers, H=4, C=64, HC=256) + mean pool + FC for MI455X (gfx1250).
//
// GEMM projections: V_WMMA_F32_16X16X4_F32 (fp32-native CDNA5 matrix op).
//   - block owns one 16-col B tile staged to LDS via GLOBAL_LOAD_ASYNC_TO_LDS
//     (ASYNCcnt DMA, ISA 08 §4), shared by 8 waves
//   - each wave computes a 64x16 strip: one B fragment feeds 4 independent
//     WMMAs per k-step (4x WMMA ILP, zero strided global B loads)
// Edge phase is L2-resident (~88MB working set << 192MB L2): one wave32 per
// edge, coalesced 1KB row gathers, f32 atomics resolved in L2.
// ---------------------------------------------------------------------------

typedef __attribute__((ext_vector_type(2))) float v2f;
typedef __attribute__((ext_vector_type(8))) float v8f;

#define HC 256
#define NHEAD 4
#define CH 64
#define NEG_SLOPE 0.2f

// ---------------------------------------------------------------------------
// Y[M,256] = X[M,256] @ W[256,256] (row-major).
// Grid: blockIdx.x = mbGroup*16 + nt.  Block = 256 threads = 8 waves.
// Block stages B[:,nt*16..nt*16+15] (256x16 f32, 16KB) into LDS.
// Wave w computes rows [ (mbGroup*8+w)*64 , +64 ) as 4 16x16 WMMA tiles.
//
// VGPR layouts (ISA 7.12.2, wave32):
//  A 16x4 f32 : lanes 0-15 -> {K=k,k+1}, lanes 16-31 -> {K=k+2,k+3}, M=lane%16
//  B 4x16 f32 : symmetric K split across lane halves, N=lane%16
//  C/D 16x16  : VGPR r: lanes 0-15 row r, lanes 16-31 row r+8, col=lane%16
// ---------------------------------------------------------------------------
__global__ __launch_bounds__(256) void gemm_f32_wmma(
    const float* __restrict__ X, const float* __restrict__ W,
    float* __restrict__ Y, int M)
{
  __shared__ float bsh[HC * 16];                 // B tile [K=256][16], 16 KB

  const int lane = threadIdx.x & 31;
  const int wave = threadIdx.x >> 5;
  const int nt = blockIdx.x & 15;                // N tile (HC/16 == 16)
  const int mb = blockIdx.x >> 4;                // 512-row block group

  // ---- stage B tile: async global->LDS DMA, 16B per lane per issue ----
  {
    const float* __restrict__ Wt = W + nt * 16;
    #pragma unroll
    for (int it = 0; it < 4; ++it) {             // 1024 x 16B chunks total
      const int idx = threadIdx.x + it * 256;    // chunk id
      const int k  = idx >> 2;                   // K row
      const int j4 = (idx & 3) * 4;              // 4-float group in row
      const unsigned lds_off =
          (unsigned)(unsigned long long)(uintptr_t)(&bsh[idx * 4]);
      const unsigned long long ga =
          (unsigned long long)(uintptr_t)(Wt + (size_t)k * HC + j4);
      asm volatile("global_load_async_to_lds_b128 %0, %1, off"
                   :: "v"(lds_off), "v"(ga) : "memory");
    }
    asm volatile("s_wait_asynccnt 0x0" ::: "memory");
  }
  __syncthreads();

  const long m0 = ((long)mb * 8 + wave) * 64;    // wave's first row
  if (m0 >= M) return;                           // wave-uniform (after barrier)

  const int hi = lane >> 4;                      // lane half
  const int lo = lane & 15;

  const float* __restrict__ Arow[4];
  #pragma unroll
  for (int t = 0; t < 4; ++t) {
    long r = m0 + t * 16 + lo;
    if (r >= M) r = M - 1;                       // clamp reads (harmless dup)
    Arow[t] = X + (size_t)r * HC;
  }

  v8f acc0 = {}, acc1 = {}, acc2 = {}, acc3 = {};
  #pragma unroll 4
  for (int k = 0; k < HC; k += 4) {
    const int kk = k + 2 * hi;
    v2f b;
    b.x = bsh[(kk + 0) * 16 + lo];               // conflict-free LDS reads
    b.y = bsh[(kk + 1) * 16 + lo];
    v2f a0, a1, a2, a3;
    a0.x = Arow[0][kk]; a0.y = Arow[0][kk + 1];
    a1.x = Arow[1][kk]; a1.y = Arow[1][kk + 1];
    a2.x = Arow[2][kk]; a2.y = Arow[2][kk + 1];
    a3.x = Arow[3][kk]; a3.y = Arow[3][kk + 1];
    acc0 = __builtin_amdgcn_wmma_f32_16x16x4_f32(false, a0, false, b, (short)0, acc0, false, false);
    acc1 = __builtin_amdgcn_wmma_f32_16x16x4_f32(false, a1, false, b, (short)0, acc1, false, false);
    acc2 = __builtin_amdgcn_wmma_f32_16x16x4_f32(false, a2, false, b, (short)0, acc2, false, false);
    acc3 = __builtin_amdgcn_wmma_f32_16x16x4_f32(false, a3, false, b, (short)0, acc3, false, false);
  }

  // ---- store: wave-uniform full-tile fast path ----
  auto store_tile = [&](const v8f& acc, int t) {
    const long base = m0 + t * 16;
    if (base >= M) return;
    float* __restrict__ Yt = Y + (size_t)base * HC + nt * 16;
    if (base + 16 <= M) {
      #pragma unroll
      for (int r = 0; r < 8; ++r)
        Yt[(size_t)(r + 8 * hi) * HC + lo] = acc[r];
    } else {
      #pragma unroll
      for (int r = 0; r < 8; ++r) {
        const int row = r + 8 * hi;
        if (base + row < M) Yt[(size_t)row * HC + lo] = acc[r];
      }
    }
  };
  store_tile(acc0, 0);
  store_tile(acc1, 1);
  store_tile(acc2, 2);
  store_tile(acc3, 3);
}

// ---------------------------------------------------------------------------
// fill helper (accumulators re-initialized every launch: harness poisons d_ws
// once and never re-poisons between graph replays)
// ---------------------------------------------------------------------------
__global__ void fill_f32(float* __restrict__ p, float v, long n)
{
  long i = (long)blockIdx.x * blockDim.x + threadIdx.x;
  if (i < n) p[i] = v;
}

// ---------------------------------------------------------------------------
// Pass 1: per-edge GATv2 score  s[e,h] = att_h . leakyrelu(xl[src]+xr[dst])
// plus segment-max into smax[dst,h] via float atomicMax.
// One wave32 per edge; each lane covers channels {lane, lane+32} per head.
// ---------------------------------------------------------------------------
__global__ __launch_bounds__(256) void edge_score_max(
    const int* __restrict__ ei, int E, int N,
    const float* __restrict__ xl, const float* __restrict__ xr,
    const float* __restrict__ att,            // [4][64]
    float* __restrict__ score,                // [E+N][4]
    float* __restrict__ smax)                 // [N][4]
{
  const int lane = threadIdx.x & 31;
  const long e = ((long)blockIdx.x * blockDim.x + threadIdx.x) >> 5;
  const long ET = (long)E + N;
  if (e >= ET) return;                        // wave-uniform
  int s, d;
  if (e < E) { s = ei[e]; d = ei[(long)E + e]; }
  else       { s = (int)(e - E); d = s; }     // self loop

  const float* __restrict__ pl = xl + (size_t)s * HC;
  const float* __restrict__ pr = xr + (size_t)d * HC;

  #pragma unroll
  for (int h = 0; h < NHEAD; ++h) {
    float acc = 0.f;
    #pragma unroll
    for (int i = 0; i < 2; ++i) {
      const int c = h * CH + lane + 32 * i;
      float v = pl[c] + pr[c];
      v = (v > 0.f) ? v : NEG_SLOPE * v;
      acc += v * att[h * CH + lane + 32 * i];
    }
    #pragma unroll
    for (int off = 16; off; off >>= 1) acc += __shfl_xor(acc, off, 32);
    if (lane == 0) {
      score[e * NHEAD + h] = acc;
      atomicMax(&smax[(size_t)d * NHEAD + h], acc);   // GLOBAL_ATOMIC_MAX_NUM_F32
    }
  }
}

// ---------------------------------------------------------------------------
// Pass 2: ex = exp(score - smax[dst]); score <- ex; denom[dst] += ex
// One thread per (edge, head).
// ---------------------------------------------------------------------------
__global__ void edge_exp_sum(
    const int* __restrict__ ei, int E, int N,
    float* __restrict__ score, const float* __restrict__ smax,
    float* __restrict__ denom)
{
  const long ET = (long)E + N;
  long idx = (long)blockIdx.x * blockDim.x + threadIdx.x;
  if (idx >= ET * NHEAD) return;
  const long e = idx >> 2;
  const int h = (int)(idx & 3);
  const int d = (e < E) ? ei[(long)E + e] : (int)(e - E);
  const float ex = __expf(score[idx] - smax[(size_t)d * NHEAD + h]);
  score[idx] = ex;
  atomicAdd(&denom[(size_t)d * NHEAD + h], ex);
}

// ---------------------------------------------------------------------------
// Pass 3: out[dst] += (ex/denom[dst]) * xl[src]   (scatter-aggregate)
// One wave32 per edge, 8 channels per lane, f32 atomics land in L2.
// ---------------------------------------------------------------------------
__global__ __launch_bounds__(256) void edge_aggregate(
    const int* __restrict__ ei, int E, int N,
    const float* __restrict__ xl,
    const float* __restrict__ score, const float* __restrict__ denom,
    float* __restrict__ out)
{
  const int lane = threadIdx.x & 31;
  const long e = ((long)blockIdx.x * blockDim.x + threadIdx.x) >> 5;
  const long ET = (long)E + N;
  if (e >= ET) return;
  int s, d;
  if (e < E) { s = ei[e]; d = ei[(long)E + e]; }
  else       { s = (int)(e - E); d = s; }

  float alpha[NHEAD];
  #pragma unroll
  for (int h = 0; h < NHEAD; ++h)
    alpha[h] = score[e * NHEAD + h] / denom[(size_t)d * NHEAD + h];

  const float* __restrict__ pl = xl + (size_t)s * HC;
  float* __restrict__ po = out + (size_t)d * HC;
  #pragma unroll
  for (int i = 0; i < 8; ++i) {
    const int c = lane + 32 * i;
    atomicAdd(&po[c], alpha[c >> 6] * pl[c]);
  }
}

// ---------------------------------------------------------------------------
// h[i] = relu(h[i] + b[i % 256])
// ---------------------------------------------------------------------------
__global__ void bias_relu(float* __restrict__ h, const float* __restrict__ b, long n)
{
  long i = (long)blockIdx.x * blockDim.x + threadIdx.x;
  if (i < n) {
    float v = h[i] + b[i & (HC - 1)];
    h[i] = v > 0.f ? v : 0.f;
  }
}

// ---------------------------------------------------------------------------
// Column sums of h2[N,256] into csum[256]
// ---------------------------------------------------------------------------
__global__ __launch_bounds__(256) void colsum_kernel(
    const float* __restrict__ h, int N, float* __restrict__ csum)
{
  const int c = threadIdx.x;                 // column 0..255
  float acc = 0.f;
  for (int r = blockIdx.x; r < N; r += gridDim.x)
    acc += h[(size_t)r * HC + c];
  atomicAdd(&csum[c], acc);
}

// ---------------------------------------------------------------------------
// out[0:2] = (csum/N) @ Wfc[256,2] + bfc   (single block, 64 threads)
// ---------------------------------------------------------------------------
__global__ void final_fc(const float* __restrict__ csum,
                         const float* __restrict__ Wfc,
                         const float* __restrict__ bfc,
                         float* __restrict__ out, float invN)
{
  __shared__ float s0[64], s1[64];
  const int t = threadIdx.x;
  float a0 = 0.f, a1 = 0.f;
  for (int k = t; k < HC; k += 64) {
    const float g = csum[k] * invN;
    a0 += g * Wfc[k * 2 + 0];
    a1 += g * Wfc[k * 2 + 1];
  }
  s0[t] = a0; s1[t] = a1;
  __syncthreads();
  if (t == 0) {
    float r0 = 0.f, r1 = 0.f;
    for (int i = 0; i < 64; ++i) { r0 += s0[i]; r1 += s1[i]; }
    out[0] = r0 + bfc[0];
    out[1] = r1 + bfc[1];
  }
}

// ---------------------------------------------------------------------------
extern "C" void kernel_launch(void* const* d_in, const int* in_sizes, int n_in,
                              void* d_out, int out_size, void* d_ws, size_t ws_size,
                              hipStream_t stream)
{
  const float* x    = (const float*)d_in[0];
  const int*   ei   = (const int*)  d_in[1];
  const float* Wl1  = (const float*)d_in[2];
  const float* Wr1  = (const float*)d_in[3];
  const float* att1 = (const float*)d_in[4];
  const float* b1   = (const float*)d_in[5];
  const float* Wl2  = (const float*)d_in[6];
  const float* Wr2  = (const float*)d_in[7];
  const float* att2 = (const float*)d_in[8];
  const float* b2   = (const float*)d_in[9];
  const float* Wfc  = (const float*)d_in[10];
  const float* bfc  = (const float*)d_in[11];
  float* out = (float*)d_out;

  const int N  = in_sizes[0] / HC;       // 20000 (IN == HC == 256)
  const int E  = in_sizes[1] / 2;        // 320000
  const long ET = (long)E + N;           // with self loops

  // ---- workspace carve (~88 MB) ----
  char* ws = (char*)d_ws;
  const size_t nodeBytes = (size_t)N * HC * sizeof(float);
  float* xl    = (float*)ws; ws += nodeBytes;
  float* xr    = (float*)ws; ws += nodeBytes;
  float* h1    = (float*)ws; ws += nodeBytes;
  float* h2    = (float*)ws; ws += nodeBytes;
  float* score = (float*)ws; ws += (size_t)ET * NHEAD * sizeof(float);
  float* smax  = (float*)ws; ws += (size_t)N * NHEAD * sizeof(float);
  float* denom = (float*)ws; ws += (size_t)N * NHEAD * sizeof(float);
  float* csum  = (float*)ws; ws += HC * sizeof(float);

  // ---- launch geometry ----
  const int mstrips64  = (N + 63) / 64;                     // 64-row wave strips
  const int gemmBlocks = 16 * ((mstrips64 + 7) / 8);        // 8 strips/block x 16 N-tiles
  const int edgeBlocks = (int)((ET + 7) / 8);               // wave per edge
  const int ehBlocks   = (int)((ET * NHEAD + 255) / 256);
  const long nodeElems = (long)N * HC;
  const int nodeBlocks = (int)((nodeElems + 255) / 256);
  const int nhBlocks   = (int)(((long)N * NHEAD + 255) / 256);
  const float NEG_INF  = -__builtin_huge_valf();

  auto run_layer = [&](const float* feat, const float* Wl, const float* Wr,
                       const float* att, const float* b, float* hout) {
    gemm_f32_wmma<<<gemmBlocks, 256, 0, stream>>>(feat, Wl, xl, N);
    gemm_f32_wmma<<<gemmBlocks, 256, 0, stream>>>(feat, Wr, xr, N);
    fill_f32<<<nhBlocks, 256, 0, stream>>>(smax, NEG_INF, (long)N * NHEAD);
    fill_f32<<<nhBlocks, 256, 0, stream>>>(denom, 0.f, (long)N * NHEAD);
    fill_f32<<<nodeBlocks, 256, 0, stream>>>(hout, 0.f, nodeElems);
    edge_score_max<<<edgeBlocks, 256, 0, stream>>>(ei, E, N, xl, xr, att, score, smax);
    edge_exp_sum<<<ehBlocks, 256, 0, stream>>>(ei, E, N, score, smax, denom);
    edge_aggregate<<<edgeBlocks, 256, 0, stream>>>(ei, E, N, xl, score, denom, hout);
    bias_relu<<<nodeBlocks, 256, 0, stream>>>(hout, b, nodeElems);
  };

  run_layer(x,  Wl1, Wr1, att1, b1, h1);   // layer 1 (+ReLU)
  run_layer(h1, Wl2, Wr2, att2, b2, h2);   // layer 2 (+ReLU)

  // mean pool + FC
  fill_f32<<<1, 256, 0, stream>>>(csum, 0.f, HC);
  colsum_kernel<<<512, 256, 0, stream>>>(h2, N, csum);
  final_fc<<<1, 64, 0, stream>>>(csum, Wfc, bfc, out, 1.0f / (float)N);
}